// EdgeFeatures_33741263077803
// MI455X (gfx1250) — compile-verified
//
#include <hip/hip_runtime.h>

typedef __attribute__((ext_vector_type(2))) float v2f;
typedef __attribute__((ext_vector_type(8))) float v8f;

#define H 256
#define BM 32
#define LDS_STRIDE 260   // 260 % 64 = 4 -> conflict-free strided fragment reads

// C = A(M x 256) @ W^T + bias, optionally += Vx[src[row]] + Vx[dst[row]]
// A row-major (K contiguous), W row-major as W[o][h] so B[k][n] = W[n][k].
// Each wave computes a 32x32 block: 2 row strips x 2 col tiles (4 WMMA / k-step,
// 2 LDS A-frag loads + 2 global B-frag loads -> 0.5 vmem per WMMA).
template <bool GATHER>
__global__ __launch_bounds__(256)
void gemm_wmma_f32(const float* __restrict__ A, int M,
                   const float* __restrict__ W,
                   const float* __restrict__ bias,
                   float* __restrict__ out,
                   const int* __restrict__ src,
                   const int* __restrict__ dst,
                   const float* __restrict__ Vx)
{
    __shared__ float smem[BM * LDS_STRIDE];

    const int tid   = threadIdx.x;
    const int lane  = tid & 31;
    const int wave  = tid >> 5;           // 0..7 : 32-column slab
    const int half  = lane >> 4;          // 0..1
    const int ml    = lane & 15;          // matrix row (A/M) or col (B/N) lane

    const int row0 = blockIdx.x * BM;

    // ---- stage A tile (BM x 256 f32) into LDS, coalesced float4 loads ----
    // 32 rows * 64 float4 = 2048 float4 -> 8 per thread
    #pragma unroll
    for (int it = 0; it < 8; ++it) {
        int i   = tid + it * 256;
        int r   = i >> 6;            // row in tile
        int c4  = i & 63;            // float4 column
        int rg  = row0 + r;
        if (rg > M - 1) rg = M - 1;  // clamp (valid memory, stores are guarded)
        const float4 v = *(const float4*)(A + (size_t)rg * H + c4 * 4);
        float* s = &smem[r * LDS_STRIDE + c4 * 4];
        s[0] = v.x; s[1] = v.y; s[2] = v.z; s[3] = v.w;
    }
    __syncthreads();

    v8f acc[2][2];
    #pragma unroll
    for (int s = 0; s < 2; ++s)
        #pragma unroll
        for (int t = 0; t < 2; ++t)
            acc[s][t] = (v8f){0.f,0.f,0.f,0.f,0.f,0.f,0.f,0.f};

    const float* smA0 = &smem[(ml)      * LDS_STRIDE];
    const float* smA1 = &smem[(16 + ml) * LDS_STRIDE];
    const float* gB0  = W + (size_t)(wave * 32 +      ml) * H;
    const float* gB1  = W + (size_t)(wave * 32 + 16 + ml) * H;

    // ---- K loop: 64 steps of K=4 ----
    for (int k0 = 0; k0 < H; k0 += 4) {
        const int ko = k0 + 2 * half;
        // A frag layout: VGPR0 = K{0|2}, VGPR1 = K{1|3} across lane halves
        v2f a0 = *(const v2f*)(smA0 + ko);
        v2f a1 = *(const v2f*)(smA1 + ko);
        v2f b0 = *(const v2f*)(gB0 + ko);
        v2f b1 = *(const v2f*)(gB1 + ko);
        acc[0][0] = __builtin_amdgcn_wmma_f32_16x16x4_f32(false, a0, false, b0,
                        (short)0, acc[0][0], false, false);
        acc[0][1] = __builtin_amdgcn_wmma_f32_16x16x4_f32(false, a0, false, b1,
                        (short)0, acc[0][1], false, false);
        acc[1][0] = __builtin_amdgcn_wmma_f32_16x16x4_f32(false, a1, false, b0,
                        (short)0, acc[1][0], false, false);
        acc[1][1] = __builtin_amdgcn_wmma_f32_16x16x4_f32(false, a1, false, b1,
                        (short)0, acc[1][1], false, false);
    }

    // ---- epilogue: bias (+ optional Vx gathers), row-guarded stores ----
    #pragma unroll
    for (int s = 0; s < 2; ++s) {
        const int rbase = row0 + s * 16 + half * 8;   // rows rbase..rbase+7
        int sg[8], dg[8];
        if (GATHER) {
            #pragma unroll
            for (int r = 0; r < 8; ++r) {
                int row = rbase + r;
                int rc  = (row < M) ? row : (M - 1);
                sg[r] = src[rc];
                dg[r] = dst[rc];
            }
        }
        #pragma unroll
        for (int t = 0; t < 2; ++t) {
            const int col = wave * 32 + t * 16 + ml;
            const float bv = bias[col];
            #pragma unroll
            for (int r = 0; r < 8; ++r) {
                const int row = rbase + r;
                if (row < M) {
                    float v = acc[s][t][r] + bv;
                    if (GATHER) {
                        v += Vx[(size_t)sg[r] * H + col];
                        v += Vx[(size_t)dg[r] * H + col];
                    }
                    out[(size_t)row * H + col] = v;
                }
            }
        }
    }
}

extern "C" void kernel_launch(void* const* d_in, const int* in_sizes, int n_in,
                              void* d_out, int out_size, void* d_ws, size_t ws_size,
                              hipStream_t stream) {
    const float* x    = (const float*)d_in[0];
    const float* e    = (const float*)d_in[1];
    const int*   ei   = (const int*)d_in[2];
    const float* Us_w = (const float*)d_in[3];
    const float* Us_b = (const float*)d_in[4];
    const float* Vs_w = (const float*)d_in[5];
    const float* Vs_b = (const float*)d_in[6];

    const int V = in_sizes[0] / H;
    const int E = in_sizes[1] / H;
    const int* src = ei;        // edge_index[0]
    const int* dst = ei + E;    // edge_index[1]

    float* Vx = (float*)d_ws;   // V*H floats (~10.24 MB), L2-resident

    dim3 block(256);

    // 1) Vx = x @ Vs_w^T + Vs_b  -> workspace
    dim3 g1((V + BM - 1) / BM);
    gemm_wmma_f32<false><<<g1, block, 0, stream>>>(
        x, V, Vs_w, Vs_b, Vx, nullptr, nullptr, nullptr);

    // 2) out = e @ Us_w^T + Us_b + Vx[src] + Vx[dst]
    dim3 g2((E + BM - 1) / BM);
    gemm_wmma_f32<true><<<g2, block, 0, stream>>>(
        e, E, Us_w, Us_b, (float*)d_out, src, dst, Vx);
}